// GINConv_69939247448309
// MI455X (gfx1250) — compile-verified
//
#include <hip/hip_runtime.h>
#include <hip/hip_bf16.h>

typedef __attribute__((ext_vector_type(2))) float v2f;
typedef __attribute__((ext_vector_type(8))) float v8f;

#define N_NODES 100000
#define N_EDGES 1600000
#define EMB 128
#define HID 256
#define BN_EPS 1e-5f

#define AGG_LD 132   // 16-row LDS tile stride (132 % 64 = 4 -> conflict-free b64 reads)
#define HID_LD 260   // 260 % 64 = 4 -> conflict-free

// ---------------------------------------------------------------- zero ws ---
__global__ __launch_bounds__(256) void zero_ws(float4* __restrict__ p, long n4) {
    long i = (long)blockIdx.x * blockDim.x + threadIdx.x;
    long stride = (long)gridDim.x * blockDim.x;
    float4 z = make_float4(0.f, 0.f, 0.f, 0.f);
    for (; i < n4; i += stride) p[i] = z;
}

// ------------------------------------------------- W1/W2 -> WMMA-B repack ---
// B fragment order: lane L supplies K rows k0+(L>>4)*2 {+0,+1} at col (L&15).
// pack[((tile*KSTEPS + ks)*32 + lane)*2 + i] = W[(ks*4 + (L>>4)*2 + i)*NCOL + tile*16 + (L&15)]
__global__ __launch_bounds__(256) void pack_w1(const float* __restrict__ W1,
                                               float* __restrict__ p1) {
    const int idx = blockIdx.x * 256 + threadIdx.x;   // 0 .. 32767
    const int i = idx & 1;
    const int lane = (idx >> 1) & 31;
    const int ks = (idx >> 6) & 31;      // 32 k-steps (K=128)
    const int t = idx >> 11;             // 16 col tiles (HID=256)
    const int k = ks * 4 + (lane >> 4) * 2 + i;
    const int col = t * 16 + (lane & 15);
    p1[idx] = W1[k * HID + col];
}

__global__ __launch_bounds__(256) void pack_w2(const float* __restrict__ W2,
                                               float* __restrict__ p2) {
    const int idx = blockIdx.x * 256 + threadIdx.x;   // 0 .. 32767
    const int i = idx & 1;
    const int lane = (idx >> 1) & 31;
    const int ks = (idx >> 6) & 63;      // 64 k-steps (K=256)
    const int t = idx >> 12;             // 8 col tiles (EMB=128)
    const int k = ks * 4 + (lane >> 4) * 2 + i;
    const int col = t * 16 + (lane & 15);
    p2[idx] = W2[k * EMB + col];
}

// ------------------------------------------- edge embed + gather + scatter ---
// One wave per edge; lane owns channels [lane*4, lane*4+4).
__global__ __launch_bounds__(256) void edge_scatter(
    const float* __restrict__ nf, const float* __restrict__ ef0,
    const float* __restrict__ ef1, const int* __restrict__ src,
    const int* __restrict__ dst, const float* __restrict__ We0,
    const float* __restrict__ We1, const float* __restrict__ be0,
    const float* __restrict__ be1, float* __restrict__ agg) {
    __shared__ float sW[9 * EMB];   // rows 0..5 = We0, rows 6..8 = We1
    __shared__ float sB[EMB];       // be0 + be1
    for (int i = threadIdx.x; i < 9 * EMB; i += 256)
        sW[i] = (i < 6 * EMB) ? We0[i] : We1[i - 6 * EMB];
    for (int i = threadIdx.x; i < EMB; i += 256)
        sB[i] = be0[i] + be1[i];
    __syncthreads();

    const int lane = threadIdx.x & 31;
    const int c0 = lane * 4;
    const int waveId = (blockIdx.x * 256 + threadIdx.x) >> 5;
    const int nWaves = (gridDim.x * 256) >> 5;

    for (int e = waveId; e < N_EDGES; e += nWaves) {
        const int s = src[e];
        const int d = dst[e];
        // lanes 0..5 fetch edge_feat_0 row, lanes 6..8 fetch edge_feat_1 row
        float f = 0.f;
        if (lane < 6)      f = ef0[(long)e * 6 + lane];
        else if (lane < 9) f = ef1[(long)e * 3 + (lane - 6)];
        float fk[9];
#pragma unroll
        for (int k = 0; k < 9; ++k) fk[k] = __shfl(f, k, 32);

        const float4 nv = *reinterpret_cast<const float4*>(nf + (long)s * EMB + c0);
        float4 acc = *reinterpret_cast<const float4*>(&sB[c0]);
        acc.x += nv.x; acc.y += nv.y; acc.z += nv.z; acc.w += nv.w;
#pragma unroll
        for (int k = 0; k < 9; ++k) {
            const float4 w = *reinterpret_cast<const float4*>(&sW[k * EMB + c0]);
            acc.x += fk[k] * w.x; acc.y += fk[k] * w.y;
            acc.z += fk[k] * w.z; acc.w += fk[k] * w.w;
        }
        float* ap = agg + (long)d * EMB + c0;
        unsafeAtomicAdd(ap + 0, acc.x);
        unsafeAtomicAdd(ap + 1, acc.y);
        unsafeAtomicAdd(ap + 2, acc.z);
        unsafeAtomicAdd(ap + 3, acc.w);
    }
}

// -------------------------------------------------- MLP via f32 WMMA tiles ---
// Block = 128 threads (4 waves) = 16 nodes. relu(agg@W1+b1)@W2+b2 -> h,
// plus per-channel sum / sum-of-squares atomics for batchnorm.
// B operands come pre-packed in fragment order: one coalesced b64 per lane/step.
__global__ __launch_bounds__(128) void mlp_wmma(
    const float* __restrict__ agg, const v2f* __restrict__ p1,
    const float* __restrict__ b1, const v2f* __restrict__ p2,
    const float* __restrict__ b2, float* __restrict__ h,
    float* __restrict__ ssum, float* __restrict__ ssq) {
    __shared__ float sA[16 * AGG_LD];
    __shared__ float sH[16 * HID_LD];

    const int node0 = blockIdx.x * 16;
    // agg tile is 16 consecutive rows x 128 cols = contiguous 2048 floats
    for (int idx = threadIdx.x; idx < 16 * 32; idx += 128) {
        const int row = idx >> 5;
        const int c4 = (idx & 31) * 4;
        const float4 v =
            *reinterpret_cast<const float4*>(agg + (long)(node0 + row) * EMB + c4);
        *reinterpret_cast<float4*>(&sA[row * AGG_LD + c4]) = v;
    }
    __syncthreads();

    const int lane = threadIdx.x & 31;
    const int wave = threadIdx.x >> 5;
    const int half = lane >> 4;   // selects K pair (A/B) and M half (C/D)
    const int l16 = lane & 15;

    // ---- stage 1: hidden = relu(agg @ W1 + b1), 16 col-tiles of 16 ----
    for (int t = wave; t < 16; t += 4) {
        const int col0 = t * 16;
        const v2f* bp = p1 + (t * 32) * 32 + lane;
        v8f c = {};
#pragma unroll
        for (int ks = 0; ks < 32; ++ks) {
            v2f a;
            const float* ap = &sA[l16 * AGG_LD + ks * 4 + half * 2];
            a.x = ap[0]; a.y = ap[1];
            const v2f b = bp[ks * 32];
            c = __builtin_amdgcn_wmma_f32_16x16x4_f32(false, a, false, b,
                                                      (short)0, c, false, false);
        }
        const float bias = b1[col0 + l16];
#pragma unroll
        for (int j = 0; j < 8; ++j) {
            float v = c[j] + bias;
            v = v > 0.f ? v : 0.f;
            sH[(half * 8 + j) * HID_LD + col0 + l16] = v;
        }
    }
    __syncthreads();

    // ---- stage 2: out = hidden @ W2 + b2, 8 col-tiles of 16 ----
    for (int t = wave; t < 8; t += 4) {
        const int col0 = t * 16;
        const v2f* bp = p2 + (t * 64) * 32 + lane;
        v8f c = {};
#pragma unroll
        for (int ks = 0; ks < 64; ++ks) {
            v2f a;
            const float* ap = &sH[l16 * HID_LD + ks * 4 + half * 2];
            a.x = ap[0]; a.y = ap[1];
            const v2f b = bp[ks * 32];
            c = __builtin_amdgcn_wmma_f32_16x16x4_f32(false, a, false, b,
                                                      (short)0, c, false, false);
        }
        const float bias = b2[col0 + l16];
        float s = 0.f, s2 = 0.f;
#pragma unroll
        for (int j = 0; j < 8; ++j) {
            const float v = c[j] + bias;
            h[(long)(node0 + half * 8 + j) * EMB + col0 + l16] = v;
            s += v;
            s2 += v * v;
        }
        // fold the two M-halves (lane and lane+16 share the same column)
        const float so = __shfl_down(s, 16, 32);
        const float s2o = __shfl_down(s2, 16, 32);
        if (half == 0) {
            unsafeAtomicAdd(&ssum[col0 + l16], s + so);
            unsafeAtomicAdd(&ssq[col0 + l16], s2 + s2o);
        }
    }
}

// ------------------------------------------------ batchnorm: scale / shift ---
__global__ void bn_finalize(const float* __restrict__ ssum,
                            const float* __restrict__ ssq,
                            const float* __restrict__ gamma,
                            const float* __restrict__ beta,
                            float* __restrict__ scale,
                            float* __restrict__ shift) {
    const int c = threadIdx.x;
    if (c < EMB) {
        const float inv_n = 1.0f / (float)N_NODES;
        const float mean = ssum[c] * inv_n;
        const float var = ssq[c] * inv_n - mean * mean;
        const float sc = gamma[c] * rsqrtf(var + BN_EPS);
        scale[c] = sc;
        shift[c] = beta[c] - mean * sc;
    }
}

__global__ __launch_bounds__(256) void bn_apply(float* __restrict__ h,
                                                const float* __restrict__ scale,
                                                const float* __restrict__ shift) {
    const long n4 = (long)N_NODES * (EMB / 4);
    long idx = (long)blockIdx.x * blockDim.x + threadIdx.x;
    if (idx < n4) {
        const int c4 = (int)(idx & (EMB / 4 - 1)) * 4;
        float4 v = reinterpret_cast<float4*>(h)[idx];
        const float4 sc = *reinterpret_cast<const float4*>(scale + c4);
        const float4 sh = *reinterpret_cast<const float4*>(shift + c4);
        v.x = v.x * sc.x + sh.x;
        v.y = v.y * sc.y + sh.y;
        v.z = v.z * sc.z + sh.z;
        v.w = v.w * sc.w + sh.w;
        reinterpret_cast<float4*>(h)[idx] = v;
    }
}

// ---------------------------------------------------------------- launch ----
extern "C" void kernel_launch(void* const* d_in, const int* in_sizes, int n_in,
                              void* d_out, int out_size, void* d_ws, size_t ws_size,
                              hipStream_t stream) {
    const float* nf   = (const float*)d_in[0];
    const float* ef0  = (const float*)d_in[1];
    const float* ef1  = (const float*)d_in[2];
    const int*   src  = (const int*)d_in[3];
    const int*   dst  = (const int*)d_in[4];
    const float* We0  = (const float*)d_in[5];
    const float* be0  = (const float*)d_in[6];
    const float* We1  = (const float*)d_in[7];
    const float* be1  = (const float*)d_in[8];
    const float* W1   = (const float*)d_in[9];
    const float* b1   = (const float*)d_in[10];
    const float* W2   = (const float*)d_in[11];
    const float* b2   = (const float*)d_in[12];
    const float* gamma = (const float*)d_in[13];
    const float* beta  = (const float*)d_in[14];
    float* out = (float*)d_out;

    float* ws    = (float*)d_ws;
    float* agg   = ws;                          // N*128 floats (51.2 MB)
    float* ssum  = ws + (long)N_NODES * EMB;    // 128
    float* ssq   = ssum + EMB;                  // 128
    float* scale = ssq + EMB;                   // 128
    float* shift = scale + EMB;                 // 128
    float* p1    = shift + EMB;                 // 32768 floats (WMMA-packed W1)
    float* p2    = p1 + 32768;                  // 32768 floats (WMMA-packed W2)

    // zero agg + stats (12,800,256 floats -> 3,200,064 float4)
    const long n4zero = ((long)N_NODES * EMB + 2 * EMB) / 4;
    zero_ws<<<(int)((n4zero + 255) / 256), 256, 0, stream>>>((float4*)ws, n4zero);

    pack_w1<<<128, 256, 0, stream>>>(W1, p1);
    pack_w2<<<128, 256, 0, stream>>>(W2, p2);

    edge_scatter<<<4096, 256, 0, stream>>>(nf, ef0, ef1, src, dst,
                                           We0, We1, be0, be1, agg);

    mlp_wmma<<<N_NODES / 16, 128, 0, stream>>>(agg, (const v2f*)p1, b1,
                                               (const v2f*)p2, b2,
                                               out, ssum, ssq);

    bn_finalize<<<1, 128, 0, stream>>>(ssum, ssq, gamma, beta, scale, shift);

    bn_apply<<<(int)(((long)N_NODES * (EMB / 4) + 255) / 256), 256, 0, stream>>>(
        out, scale, shift);
}